// SWGATLayer_83992380440762
// MI455X (gfx1250) — compile-verified
//
#include <hip/hip_runtime.h>
#include <math.h>

typedef __attribute__((ext_vector_type(2))) float v2f;
typedef __attribute__((ext_vector_type(8))) float v8f;

#define IN_DIM 128
#define OUT_DIM 64

// ---------------------------------------------------------------------------
// Init: m = -inf, den = 0, out = 0  (harness poisons buffers; re-init per call)
// ---------------------------------------------------------------------------
__global__ void init_buffers(float* __restrict__ m, float* __restrict__ den,
                             float* __restrict__ out, int NW) {
    int i = blockIdx.x * blockDim.x + threadIdx.x;
    int total = NW * OUT_DIM;
    if (i < total) out[i] = 0.0f;
    if (i < NW) { m[i] = -INFINITY; den[i] = 0.0f; }
}

// ---------------------------------------------------------------------------
// Fused: z[NS,64] = h[NS,128] @ W[128,64]  AND  s[i] = dot(z[i,:], wa[0:64]).
// One wave32 computes a full 16x64 slab: A (16x4 f32) is loaded once per
// K-step and fed into 4 independent V_WMMA_F32_16X16X4_F32 accumulation
// chains (one per 16-col N-tile). Epilogue reduces s via shfl_xor butterfly.
//
// A 16x4 layout: lanes 0-15 -> rows M=l, v[0]=K+0, v[1]=K+1;
//                lanes 16-31 -> rows M=l, v[0]=K+2, v[1]=K+3.
// B 4x16 mirrors with N=l. C/D: vgpr g -> row g + 8*(lane>=16), col = l.
// ---------------------------------------------------------------------------
__global__ __launch_bounds__(256) void gemm_zs(const float* __restrict__ h,
                                               const float* __restrict__ W,
                                               const float* __restrict__ w_attn,
                                               float* __restrict__ z,
                                               float* __restrict__ s, int NS) {
    const int waves_per_block = blockDim.x >> 5;
    const int wave = blockIdx.x * waves_per_block + (threadIdx.x >> 5);
    const int lane = threadIdx.x & 31;
    const int m0 = wave * 16;
    if (m0 >= NS) return;                          // whole-wave uniform exit

    const int half = lane >> 4;                    // 0 -> K+0/1, 1 -> K+2/3
    const int l = lane & 15;
    int rowA = m0 + l;
    if (rowA >= NS) rowA = NS - 1;                 // clamp: valid addr, no EXEC churn

    v8f acc0 = {}, acc1 = {}, acc2 = {}, acc3 = {};
    const float* ap = h + (size_t)rowA * IN_DIM + half * 2;
#pragma unroll 4
    for (int k = 0; k < IN_DIM; k += 4) {
        v2f a = *(const v2f*)(ap + k);
        const float* bp = W + (size_t)(k + half * 2) * OUT_DIM + l;
        v2f b0, b1, b2, b3;
        b0.x = bp[0];      b0.y = bp[OUT_DIM];
        b1.x = bp[16];     b1.y = bp[OUT_DIM + 16];
        b2.x = bp[32];     b2.y = bp[OUT_DIM + 32];
        b3.x = bp[48];     b3.y = bp[OUT_DIM + 48];
        acc0 = __builtin_amdgcn_wmma_f32_16x16x4_f32(false, a, false, b0, (short)0, acc0, false, false);
        acc1 = __builtin_amdgcn_wmma_f32_16x16x4_f32(false, a, false, b1, (short)0, acc1, false, false);
        acc2 = __builtin_amdgcn_wmma_f32_16x16x4_f32(false, a, false, b2, (short)0, acc2, false, false);
        acc3 = __builtin_amdgcn_wmma_f32_16x16x4_f32(false, a, false, b3, (short)0, acc3, false, false);
    }

    // ---- store z slab: vgpr g -> row m0 + g + half*8, cols l, l+16, l+32, l+48
#pragma unroll
    for (int g = 0; g < 8; ++g) {
        int row = m0 + g + half * 8;
        if (row < NS) {
            float* zr = z + (size_t)row * OUT_DIM + l;
            zr[0]  = acc0[g];
            zr[16] = acc1[g];
            zr[32] = acc2[g];
            zr[48] = acc3[g];
        }
    }

    // ---- fused s = z @ wa_src: per-lane partials, butterfly over 16-lane half
    float wa0 = w_attn[l];
    float wa1 = w_attn[16 + l];
    float wa2 = w_attn[32 + l];
    float wa3 = w_attn[48 + l];
    float p[8];
#pragma unroll
    for (int g = 0; g < 8; ++g)
        p[g] = acc0[g] * wa0 + acc1[g] * wa1 + acc2[g] * wa2 + acc3[g] * wa3;
#pragma unroll
    for (int off = 1; off < 16; off <<= 1) {
#pragma unroll
        for (int g = 0; g < 8; ++g)
            p[g] += __shfl_xor(p[g], off, 32);
    }
    if (l < 8) {
        int row = m0 + l + half * 8;
        if (row < NS) s[row] = p[l];
    }
}

// Float atomic max via int/uint bit ordering (init must be -inf).
__device__ __forceinline__ void atomicMaxF(float* addr, float val) {
    if (val >= 0.0f)
        atomicMax((int*)addr, __float_as_int(val));
    else
        atomicMin((unsigned int*)addr, __float_as_uint(val));
}

// ---------------------------------------------------------------------------
// Pass 1: e = weight * leaky_relu(s[src]); segment max into m[dst]
// ---------------------------------------------------------------------------
__global__ __launch_bounds__(256) void edge_logits(const float* __restrict__ s,
                                                   const float* __restrict__ weight,
                                                   const int* __restrict__ src,
                                                   const int* __restrict__ dst,
                                                   float* __restrict__ e,
                                                   float* __restrict__ m, int E) {
    int i = blockIdx.x * blockDim.x + threadIdx.x;
    if (i >= E) return;
    float sv = s[src[i]];
    float lv = (sv > 0.0f) ? sv : 0.01f * sv;   // jax default leaky_relu slope
    float ev = weight[i] * lv;
    e[i] = ev;
    atomicMaxF(m + dst[i], ev);
}

// ---------------------------------------------------------------------------
// Pass 2: ex = exp(e - m[dst]) (in place over e); den[dst] += ex
// ---------------------------------------------------------------------------
__global__ __launch_bounds__(256) void edge_exp_sum(const int* __restrict__ dst,
                                                    const float* __restrict__ m,
                                                    float* __restrict__ e,
                                                    float* __restrict__ den, int E) {
    int i = blockIdx.x * blockDim.x + threadIdx.x;
    if (i >= E) return;
    int d = dst[i];
    float ex = expf(e[i] - m[d]);
    e[i] = ex;
    atomicAdd(den + d, ex);
}

// ---------------------------------------------------------------------------
// Pass 3: out[dst,:] += (ex/den[dst]) * z[src,:]
// 16 threads per edge -> coalesced 256B gather of z row; 4 f32 atomics each.
// ---------------------------------------------------------------------------
__global__ __launch_bounds__(256) void edge_scatter(const int* __restrict__ src,
                                                    const int* __restrict__ dst,
                                                    const float* __restrict__ ex,
                                                    const float* __restrict__ den,
                                                    const float* __restrict__ z,
                                                    float* __restrict__ out, int E) {
    long long t = (long long)blockIdx.x * blockDim.x + threadIdx.x;
    long long edge = t >> 4;
    if (edge >= E) return;
    int c4 = (int)(t & 15) * 4;
    int sIdx = src[edge];
    int dIdx = dst[edge];
    float alpha = ex[edge] / den[dIdx];
    float4 zv = *(const float4*)(z + (size_t)sIdx * OUT_DIM + c4);
    float* op = out + (size_t)dIdx * OUT_DIM + c4;
    atomicAdd(op + 0, alpha * zv.x);
    atomicAdd(op + 1, alpha * zv.y);
    atomicAdd(op + 2, alpha * zv.z);
    atomicAdd(op + 3, alpha * zv.w);
}

extern "C" void kernel_launch(void* const* d_in, const int* in_sizes, int n_in,
                              void* d_out, int out_size, void* d_ws, size_t ws_size,
                              hipStream_t stream) {
    (void)n_in; (void)ws_size;
    const float* h      = (const float*)d_in[0];
    const float* W_fc   = (const float*)d_in[1];
    const float* w_attn = (const float*)d_in[2];
    const float* weight = (const float*)d_in[3];
    const int*   src    = (const int*)d_in[4];
    const int*   dst    = (const int*)d_in[5];
    float* out = (float*)d_out;

    const int NS = in_sizes[0] / IN_DIM;
    const int E  = in_sizes[3];
    const int NW = out_size / OUT_DIM;

    // Workspace layout (floats): z[NS*64] | s[NS] | e[E] | m[NW] | den[NW]
    float* z   = (float*)d_ws;
    float* s   = z + (size_t)NS * OUT_DIM;
    float* e   = s + NS;
    float* m   = e + E;
    float* den = m + NW;

    // init m/den/out
    {
        int total = NW * OUT_DIM;
        int blocks = (total + 255) / 256;
        init_buffers<<<blocks, 256, 0, stream>>>(m, den, out, NW);
    }
    // z = h @ W_fc  (WMMA) fused with s = z @ wa_src
    {
        int waves = (NS + 15) / 16;     // one wave per 16-row slab
        int blocks = (waves + 7) / 8;   // 8 waves per 256-thread block
        gemm_zs<<<blocks, 256, 0, stream>>>(h, W_fc, w_attn, z, s, NS);
    }
    // edge pass 1: logits + segment max
    edge_logits<<<(E + 255) / 256, 256, 0, stream>>>(s, weight, src, dst, e, m, E);
    // edge pass 2: exp + segment sum
    edge_exp_sum<<<(E + 255) / 256, 256, 0, stream>>>(dst, m, e, den, E);
    // edge pass 3: weighted scatter
    {
        long long threads = (long long)E * 16;
        long long blocks = (threads + 255) / 256;
        edge_scatter<<<(unsigned)blocks, 256, 0, stream>>>(src, dst, e, den, z, out, E);
    }
}